// LimbDarkLightCurve_55370718380097
// MI455X (gfx1250) — compile-verified
//
#include <hip/hip_runtime.h>
#include <math.h>
#include <stdint.h>

// ---------------------------------------------------------------------------
// Limb-darkened transit light curve, K = 2^21 samples, order-10 Gauss-Legendre
// quadrature. Transcendental-VALU-bound elementwise kernel:
//   bytes/sample = 12 -> ~1.1 us at 23.3 TB/s; ~450 f32 ops/sample -> several
//   us of VALU work => compute-bound. WMMA shapes cannot express the
//   data-dependent per-sample quadrature without >10x wasted MACs, so the
//   CDNA5-specific paths used are the async global->LDS copy (ASYNCcnt path)
//   and global prefetch; the rest is tightly unrolled f32 VALU.
// ---------------------------------------------------------------------------

namespace {

constexpr int ORDER = 10;
constexpr int TPB = 256;        // threads per block
constexpr int SPT = 4;          // samples per thread (float4)

__device__ __forceinline__ float quad_intensity(float u1, float u2, float rho) {
  // I(mu) = 1 - u1*(1-mu) - u2*(1-mu)^2 ; mu = sqrt(clip(1-rho^2, 1e-12))
  // Argument is clamped to [1e-12, 1] -> raw v_sqrt_f32 (1 ulp) is exact
  // enough; skip the ocml denormal-rescale fixup.
  float mu = __builtin_amdgcn_sqrtf(fmaxf(fmaf(-rho, rho, 1.0f), 1e-12f));
  float omm = 1.0f - mu;
  return 1.0f - omm * (u1 + u2 * omm);
}

// ~0.5-ulp reciprocal: v_rcp_f32 + one Newton-Raphson step.
__device__ __forceinline__ float fast_rcp(float d) {
  float r = __builtin_amdgcn_rcpf(d);
  r = fmaf(r, fmaf(-d, r, 1.0f), r);
  return r;
}

__global__ __launch_bounds__(TPB) void limbdark_kernel(
    const float* __restrict__ u, const float* __restrict__ b,
    const float* __restrict__ rr, const float* __restrict__ z,
    float* __restrict__ out, int K) {
  // Gauss-Legendre order-10 nodes/weights (np.polynomial.legendre.leggauss(10))
  constexpr float xs[ORDER] = {
      -0.9739065285171717f, -0.8650633666889845f, -0.6794095682990244f,
      -0.4333953941292472f, -0.1488743389816312f,  0.1488743389816312f,
       0.4333953941292472f,  0.6794095682990244f,  0.8650633666889845f,
       0.9739065285171717f};
  constexpr float wq[ORDER] = {
      0.06667134430868814f, 0.14945134915058059f, 0.21908636251598204f,
      0.26926671930999635f, 0.29552422471475287f, 0.29552422471475287f,
      0.26926671930999635f, 0.21908636251598204f, 0.14945134915058059f,
      0.06667134430868814f};
  // Annulus weights with the alpha = 2*acos(...) factor folded in.
  constexpr float wq2[ORDER] = {
      2.0f * 0.06667134430868814f, 2.0f * 0.14945134915058059f,
      2.0f * 0.21908636251598204f, 2.0f * 0.26926671930999635f,
      2.0f * 0.29552422471475287f, 2.0f * 0.29552422471475287f,
      2.0f * 0.26926671930999635f, 2.0f * 0.21908636251598204f,
      2.0f * 0.14945134915058059f, 2.0f * 0.06667134430868814f};

  __shared__ float tileB[TPB * SPT];

  const int t = threadIdx.x;
  const long long base = ((long long)blockIdx.x * TPB + t) * SPT;
  if (base >= (long long)K) return;

  const bool full = (base + (SPT - 1)) < (long long)K;

  // ---- CDNA5 async global->LDS staging of this thread's 4 b-values ----
  if (full) {
    const float* gsrc = b + base;
    uint32_t lds_dst = (uint32_t)(uintptr_t)(&tileB[t * SPT]);
    asm volatile("global_load_async_to_lds_b128 %0, %1, off"
                 :: "v"(lds_dst), "v"(gsrc)
                 : "memory");
    // Prefetch z while the async copy and uniform setup are in flight.
    __builtin_prefetch(z + base, 0, 3);
  }

  // Uniform setup (scalarized by the compiler).
  const float r  = rr[0];
  const float u1 = u[0];
  const float u2 = u[1];
  // norm = pi * (1 - u1*2/(2*3) - u2*2/(3*4))
  const float norm =
      3.14159265358979323846f * (1.0f - u1 * (1.0f / 3.0f) - u2 * (1.0f / 6.0f));
  const float inv_norm = 1.0f / norm;
  const float r2 = r * r;

  float bv[SPT], zv[SPT];
  if (full) {
    asm volatile("s_wait_asynccnt 0" ::: "memory");
    float4 b4 = *reinterpret_cast<const float4*>(&tileB[t * SPT]);
    float4 z4 = *reinterpret_cast<const float4*>(z + base);
    bv[0] = b4.x; bv[1] = b4.y; bv[2] = b4.z; bv[3] = b4.w;
    zv[0] = z4.x; zv[1] = z4.y; zv[2] = z4.z; zv[3] = z4.w;
  } else {
    #pragma unroll
    for (int i = 0; i < SPT; ++i) {
      long long idx = base + i;
      bv[i] = (idx < (long long)K) ? b[idx] : 0.0f;
      zv[i] = (idx < (long long)K) ? z[idx] : 0.0f;
    }
  }

  float res[SPT];
  #pragma unroll
  for (int i = 0; i < SPT; ++i) {
    const float bi = fabsf(bv[i]);

    // ---- fully occulted core: rho in [0, clip(r-b,0,1)] ----
    const float core_hi = fminf(fmaxf(r - bi, 0.0f), 1.0f);
    const float half_c = 0.5f * core_hi;
    float sum_c = 0.0f;
    #pragma unroll
    for (int j = 0; j < ORDER; ++j) {
      const float rho = half_c * (xs[j] + 1.0f);
      sum_c = fmaf(wq[j] * rho, quad_intensity(u1, u2, rho), sum_c);
    }
    const float f_core = half_c * 6.28318530717958647692f * sum_c;

    // ---- partial annulus: rho in [clip(|b-r|,0,1), clip(b+r,0,1)] ----
    const float lo = fminf(fabsf(bi - r), 1.0f);
    const float hi = fminf(bi + r, 1.0f);
    const float width = fmaxf(hi - lo, 0.0f);
    const float half_w = 0.5f * width;
    const float b2mr2 = fmaf(bi, bi, -r2);
    const float twob = bi + bi;
    float sum_a = 0.0f;
    #pragma unroll
    for (int j = 0; j < ORDER; ++j) {
      const float rho = fmaf(half_w, xs[j] + 1.0f, lo);
      const float denom = twob * rho;
      const bool safe = denom > 1e-12f;
      // where(safe, denom, 1.0) BEFORE the reciprocal: everything stays finite.
      float ca = fmaf(rho, rho, b2mr2) * fast_rcp(safe ? denom : 1.0f);
      ca = fminf(fmaxf(ca, -0.999999f), 0.999999f);
      const float aco = safe ? acosf(ca) : 0.0f;  // alpha = 2*aco (2 in wq2)
      sum_a = fmaf(wq2[j] * rho, quad_intensity(u1, u2, rho) * aco, sum_a);
    }
    const float f_ann = half_w * sum_a;

    const float lc = -(f_core + f_ann) * inv_norm;
    res[i] = (zv[i] > 0.0f) ? lc : 0.0f;
  }

  if (full) {
    *reinterpret_cast<float4*>(out + base) =
        make_float4(res[0], res[1], res[2], res[3]);
  } else {
    #pragma unroll
    for (int i = 0; i < SPT; ++i) {
      long long idx = base + i;
      if (idx < (long long)K) out[idx] = res[i];
    }
  }
}

}  // namespace

extern "C" void kernel_launch(void* const* d_in, const int* in_sizes, int n_in,
                              void* d_out, int out_size, void* d_ws, size_t ws_size,
                              hipStream_t stream) {
  (void)n_in; (void)d_ws; (void)ws_size;
  // setup_inputs order: u (2 x f32), b (K x f32), r (1 x f32), z (K x f32),
  // order (1 x i32, statically 10 -> nodes hardcoded).
  const float* u = (const float*)d_in[0];
  const float* b = (const float*)d_in[1];
  const float* r = (const float*)d_in[2];
  const float* z = (const float*)d_in[3];
  float* out = (float*)d_out;

  const int K = out_size;  // == in_sizes[1]
  const int groups = (K + SPT - 1) / SPT;
  const int blocks = (groups + TPB - 1) / TPB;
  limbdark_kernel<<<blocks, TPB, 0, stream>>>(u, b, r, z, out, K);
}